// compute_G_91070486544608
// MI455X (gfx1250) — compile-verified
//
#include <hip/hip_runtime.h>
#include <hip/hip_bf16.h>

typedef __attribute__((ext_vector_type(2))) float v2f;
typedef __attribute__((ext_vector_type(8))) float v8f;

#define N_DIM 8192
#define E_DIM 4096
#define BM 128
#define BN 128
#define BK 16
#define BMP 136   // padded LDS leading dim: 2*136 % 64 == 16 -> half-wave groups hit disjoint banks
#define BNP 136

__global__ __launch_bounds__(256)
void hgnn_wmma_f32_kernel(const float* __restrict__ A,     // [N, E] row-major
                          const float* __restrict__ Binv,  // [E, N] row-major
                          const float* __restrict__ W,     // [E]
                          float* __restrict__ G)           // [N, N]
{
    __shared__ float As[2][BK * BMP];  // As[buf][k*BMP + m]  (transposed tile)
    __shared__ float Bs[2][BK * BNP];  // Bs[buf][k*BNP + n]  (W pre-folded)

    const int tid  = threadIdx.x;
    const int lane = tid & 31;
    const int wave = tid >> 5;      // 0..7
    const int wm   = wave >> 1;     // 0..3 : wave row (32 rows each)
    const int wn   = wave & 1;      // 0..1 : wave col (64 cols each)
    const int row  = lane & 15;     // N (or M) index within 16x16 tile
    const int kh   = lane >> 4;     // half-wave selector (K pair / M half)

    const int bm = blockIdx.y * BM;
    const int bn = blockIdx.x * BN;

    // ---- global->LDS staging assignments ----
    // A: 2 threads per row; thread covers k = akb+8i+{0..3}, i=0..1
    const int am  = tid >> 1;          // 0..127
    const int akb = (tid & 1) * 4;     // 0 or 4
    // B: 16 threads per k-row; thread covers n = bnb+64i+{0..3}, i=0..1
    const int bk  = tid >> 4;          // 0..15
    const int bnb = (tid & 15) * 4;    // 0..60

    const float* Aptr = A    + (size_t)(bm + am) * E_DIM + akb;
    const float* Bptr = Binv + (size_t)bk * N_DIM + (bn + bnb);

    float4 aReg[2], bReg[2];
    float  wReg = 0.0f;

    v8f acc[2][4];
    const v8f vzero = {0.f, 0.f, 0.f, 0.f, 0.f, 0.f, 0.f, 0.f};
#pragma unroll
    for (int tm = 0; tm < 2; ++tm)
#pragma unroll
        for (int tn = 0; tn < 4; ++tn)
            acc[tm][tn] = vzero;

    auto loadGlobal = [&](int k0) {
        aReg[0] = *(const float4*)(Aptr + k0);
        aReg[1] = *(const float4*)(Aptr + k0 + 8);
        wReg = W[k0 + bk];
        const float* bp = Bptr + (size_t)k0 * N_DIM;
        bReg[0] = *(const float4*)(bp);
        bReg[1] = *(const float4*)(bp + 64);
    };

    auto storeLDS = [&](int buf) {
#pragma unroll
        for (int i = 0; i < 2; ++i) {
            const int k = akb + 8 * i;
            const float4 v = aReg[i];
            As[buf][(k + 0) * BMP + am] = v.x;
            As[buf][(k + 1) * BMP + am] = v.y;
            As[buf][(k + 2) * BMP + am] = v.z;
            As[buf][(k + 3) * BMP + am] = v.w;
        }
#pragma unroll
        for (int i = 0; i < 2; ++i) {
            float4 v = bReg[i];
            v.x *= wReg; v.y *= wReg; v.z *= wReg; v.w *= wReg;
            *(float4*)&Bs[buf][bk * BNP + bnb + 64 * i] = v;
        }
    };

    // prologue: stage tile 0
    loadGlobal(0);
    storeLDS(0);
    __syncthreads();

    const int numTiles = E_DIM / BK;   // 256
    for (int t = 0; t < numTiles; ++t) {
        const int buf = t & 1;
        if (t + 1 < numTiles) loadGlobal((t + 1) * BK);  // overlap HBM with compute

        // ---- compute: 4 k-steps x (2x4) wmma tiles ----
#pragma unroll
        for (int ks = 0; ks < BK / 4; ++ks) {
            const int k0 = ks * 4 + 2 * kh;   // this half-wave's K pair (k0, k0+1)
            v2f aF[2], bF[4];
#pragma unroll
            for (int tm = 0; tm < 2; ++tm) {
                const int m = wm * 32 + tm * 16 + row;
                v2f a;
                a.x = As[buf][(k0 + 0) * BMP + m];
                a.y = As[buf][(k0 + 1) * BMP + m];
                aF[tm] = a;
            }
#pragma unroll
            for (int tn = 0; tn < 4; ++tn) {
                const int n = wn * 64 + tn * 16 + row;
                v2f b;
                b.x = Bs[buf][(k0 + 0) * BNP + n];
                b.y = Bs[buf][(k0 + 1) * BNP + n];
                bF[tn] = b;
            }
#pragma unroll
            for (int tm = 0; tm < 2; ++tm)
#pragma unroll
                for (int tn = 0; tn < 4; ++tn)
                    acc[tm][tn] = __builtin_amdgcn_wmma_f32_16x16x4_f32(
                        /*neg_a=*/false, aF[tm],
                        /*neg_b=*/false, bF[tn],
                        /*c_mod=*/(short)0, acc[tm][tn],
                        /*reuse_a=*/false, /*reuse_b=*/false);
        }

        if (t + 1 < numTiles) storeLDS(buf ^ 1);  // other buffer: safe pre-barrier
        __syncthreads();
    }

    // ---- epilogue: C/D layout -> coalesced row-segment stores ----
#pragma unroll
    for (int tm = 0; tm < 2; ++tm) {
        const int gm0 = bm + wm * 32 + tm * 16 + kh * 8;  // VGPR i holds M = 8*kh + i
#pragma unroll
        for (int tn = 0; tn < 4; ++tn) {
            const int gn = bn + wn * 64 + tn * 16 + row;
#pragma unroll
            for (int i = 0; i < 8; ++i)
                G[(size_t)(gm0 + i) * N_DIM + gn] = acc[tm][tn][i];
        }
    }
}

extern "C" void kernel_launch(void* const* d_in, const int* in_sizes, int n_in,
                              void* d_out, int out_size, void* d_ws, size_t ws_size,
                              hipStream_t stream) {
    (void)in_sizes; (void)n_in; (void)d_ws; (void)ws_size; (void)out_size;
    const float* A    = (const float*)d_in[0];  // DV2_H        [N, E]
    const float* Binv = (const float*)d_in[1];  // invDE_HT_DV2 [E, N]
    const float* W    = (const float*)d_in[2];  // W            [E]
    float* G          = (float*)d_out;          // [N, N]

    dim3 grid(N_DIM / BN, N_DIM / BM);  // 64 x 64
    dim3 block(256);
    hgnn_wmma_f32_kernel<<<grid, block, 0, stream>>>(A, Binv, W, G);
}